// OD2Score_57346403336462
// MI455X (gfx1250) — compile-verified
//
#include <hip/hip_runtime.h>
#include <stdint.h>

// ---------------- problem constants (match reference) ----------------
#define BATCH   16
#define NROWS   25200
#define NCH     85
#define NCLS    80
#define KSEL    512
#define CONF_T  0.01f
#define IOU_T   0.5f
#define MAX_WH  4096.0f

typedef __attribute__((ext_vector_type(2))) float v2f;
typedef __attribute__((ext_vector_type(8))) float v8f;

// Map float -> order-preserving unsigned (ascending), and back.
__device__ __forceinline__ unsigned f2ord(float f) {
    unsigned b = __float_as_uint(f);
    return (b & 0x80000000u) ? ~b : (b | 0x80000000u);
}
__device__ __forceinline__ float ord2f(unsigned u) {
    return (u & 0x80000000u) ? __uint_as_float(u & 0x7FFFFFFFu)
                             : __uint_as_float(~u);
}

// ============================================================================
// Kernel 1: streaming score pass (bandwidth-bound: 137 MB once through HBM).
// Each block processes 2 tiles of 64 rows; rows staged through LDS so global
// reads are fully coalesced; next tile is software-prefetched
// (global_prefetch_b8). Emits one unique 64-bit sortable key per row:
//   key = ( ord(valid ? conf : -1) << 32 ) | (0xFFFFFFFF - row)
// so descending 64-bit order == top_k's (value desc, index asc) order.
// ============================================================================
__global__ __launch_bounds__(256) void score_kernel(
        const float* __restrict__ x, unsigned long long* __restrict__ keys) {
    __shared__ float sm[64 * NCH];     // 21.76 KB row tile
    __shared__ float pm[4 * 64];       // partial class maxima
    const int b   = blockIdx.y;
    const int tid = threadIdx.x;

    for (int sub = 0; sub < 2; ++sub) {
        const int base = (blockIdx.x * 2 + sub) * 64;
        if (base >= NROWS) break;
        const int nrow = min(64, NROWS - base);
        const size_t goff = ((size_t)b * NROWS + base) * NCH;
        const int lim = nrow * NCH;

        for (int i = tid; i < 64 * NCH; i += 256)
            sm[i] = (i < lim) ? x[goff + i] : 0.0f;

        // gfx1250 software prefetch of the next tile while this one reduces
        if (sub == 0 && base + 64 < NROWS)
            __builtin_prefetch(&x[goff + (size_t)64 * NCH + (size_t)tid * 32], 0, 0);
        __syncthreads();

        // 4 threads cooperate per row: each scans 20 of the 80 class channels
        {
            const int r = tid & 63, part = tid >> 6;
            float mx = -1e30f;
            const int c0 = 5 + part * 20;
            #pragma unroll
            for (int c = 0; c < 20; ++c) mx = fmaxf(mx, sm[r * NCH + c0 + c]);
            pm[part * 64 + r] = mx;
        }
        __syncthreads();

        if (tid < 64 && tid < nrow) {
            const float m = fmaxf(fmaxf(pm[tid], pm[64 + tid]),
                                  fmaxf(pm[128 + tid], pm[192 + tid]));
            const float obj  = sm[tid * NCH + 4];
            const float conf = obj * m;                       // == max(cls*obj)
            const bool  valid = (obj > CONF_T) && (conf > CONF_T);
            const float mv = valid ? conf : -1.0f;
            const unsigned row = (unsigned)(base + tid);
            const unsigned long long key =
                ((unsigned long long)f2ord(mv) << 32) |
                (unsigned long long)(0xFFFFFFFFu - row);
            keys[(size_t)b * NROWS + base + tid] = key;
        }
        __syncthreads();   // before next sub-tile reuses sm
    }
}

// ============================================================================
// Kernel 2: per-batch exact top-512. MSB radix-select of the 512th-largest
// 64-bit key (keys are unique), gather the 512 keys >= T, bitonic sort
// descending in LDS, write sorted keys.
// ============================================================================
__global__ __launch_bounds__(256) void select_kernel(
        const unsigned long long* __restrict__ keys,
        unsigned long long* __restrict__ sorted) {
    __shared__ unsigned hist[256];
    __shared__ unsigned long long sel[KSEL];
    __shared__ unsigned long long s_prefix;
    __shared__ int s_remain;
    __shared__ unsigned s_cnt;

    const int b   = blockIdx.x;
    const int tid = threadIdx.x;
    const unsigned long long* kb = keys + (size_t)b * NROWS;

    if (tid == 0) { s_prefix = 0ull; s_remain = KSEL; s_cnt = 0u; }

    for (int round = 0; round < 8; ++round) {
        __syncthreads();
        hist[tid] = 0u;
        __syncthreads();
        const int shift = 56 - 8 * round;
        const unsigned long long pfx = s_prefix;
        for (int n = tid; n < NROWS; n += 256) {
            const unsigned long long k = kb[n];
            const bool ok = (round == 0) || ((k >> (shift + 8)) == pfx);
            if (ok) atomicAdd(&hist[(unsigned)((k >> shift) & 0xFFull)], 1u);
        }
        __syncthreads();
        if (tid == 0) {
            const int rk = s_remain;
            int acc = 0, bsel = 0;
            for (int bin = 255; bin >= 0; --bin) {
                const int c = (int)hist[bin];
                if (acc + c >= rk) { bsel = bin; break; }
                acc += c;
            }
            s_prefix = (s_prefix << 8) | (unsigned long long)bsel;
            s_remain = rk - acc;
        }
    }
    __syncthreads();

    const unsigned long long T = s_prefix;   // exactly 512 keys are >= T
    for (int n = tid; n < NROWS; n += 256) {
        const unsigned long long k = kb[n];
        if (k >= T) {
            const unsigned p = atomicAdd(&s_cnt, 1u);
            if (p < KSEL) sel[p] = k;
        }
    }
    __syncthreads();

    // bitonic sort, overall descending
    for (unsigned size = 2; size <= KSEL; size <<= 1) {
        for (unsigned stride = size >> 1; stride > 0; stride >>= 1) {
            __syncthreads();
            for (unsigned t = tid; t < KSEL; t += 256) {
                const unsigned p = t ^ stride;
                if (p > t) {
                    const bool descBlk = ((t & size) == 0);
                    const unsigned long long a = sel[t], c = sel[p];
                    if (descBlk ? (a < c) : (a > c)) { sel[t] = c; sel[p] = a; }
                }
            }
        }
    }
    __syncthreads();
    sorted[(size_t)b * KSEL + tid]       = sel[tid];
    sorted[(size_t)b * KSEL + tid + 256] = sel[tid + 256];
}

// ============================================================================
// Kernel 3: per-batch NMS + target scoring. One block of 512 threads
// (16 wave32s). Suppression matrix held as 512x16 uint32 bitmask in LDS.
// Pairwise (area_i + area_j) computed per 16x16 tile with
// V_WMMA_F32_16X16X4_F32 (rank-2 outer product [area,1]x[1,area]^T).
// ============================================================================
__global__ __launch_bounds__(512) void nms_kernel(
        const float* __restrict__ x, const float* __restrict__ target,
        const unsigned long long* __restrict__ sorted,
        float* __restrict__ out) {
    __shared__ float    sbx [KSEL][4];   // class-offset boxes
    __shared__ float    sraw[KSEL][4];   // raw boxes (for target IoU)
    __shared__ float    sarea[KSEL];
    __shared__ float    sscore[KSEL];
    __shared__ int      scls[KSEL];
    __shared__ unsigned smask[KSEL][16]; // suppression bitmask (32 KB)
    __shared__ unsigned skeep[16];
    __shared__ float    redv[KSEL];
    __shared__ int      redi[KSEL];

    const int b   = blockIdx.x;
    const int tid = threadIdx.x;

    // ---- decode selected rows; recompute class/boxes for just 512 rows ----
    {
        const unsigned long long key = sorted[(size_t)b * KSEL + tid];
        const unsigned row = 0xFFFFFFFFu - (unsigned)(key & 0xFFFFFFFFull);
        const float score = ord2f((unsigned)(key >> 32));
        const float* p = x + ((size_t)b * NROWS + row) * NCH;
        float best = p[5]; int bc = 0;          // first-max == jnp.argmax
        for (int c = 1; c < NCLS; ++c) {
            const float v = p[5 + c];
            if (v > best) { best = v; bc = c; }
        }
        const float cx = p[0], cy = p[1], hw = p[2] * 0.5f, hh = p[3] * 0.5f;
        const float rx1 = cx - hw, ry1 = cy - hh, rx2 = cx + hw, ry2 = cy + hh;
        const float off = (float)bc * MAX_WH;
        const float ox1 = rx1 + off, oy1 = ry1 + off;
        const float ox2 = rx2 + off, oy2 = ry2 + off;
        sraw[tid][0] = rx1; sraw[tid][1] = ry1; sraw[tid][2] = rx2; sraw[tid][3] = ry2;
        sbx [tid][0] = ox1; sbx [tid][1] = oy1; sbx [tid][2] = ox2; sbx [tid][3] = oy2;
        sarea[tid]  = (ox2 - ox1) * (oy2 - oy1);  // same math as reference (on offset boxes)
        sscore[tid] = score;
        scls[tid]   = bc;
        #pragma unroll
        for (int w = 0; w < 16; ++w) smask[tid][w] = 0u;
    }
    __syncthreads();

    // ---- build 512x512 suppression bitmask; one wave per 16x16 tile ----
    const int lane = tid & 31;
    const int wave = tid >> 5;
    // Gate: lanes 16..31 feed the (zero) K=2,3 coefficients of the 16x16x4
    // WMMA. Arithmetic gating (multiply by 0/1) keeps the math correct under
    // any predication strategy the compiler chooses.
    const float gate = (lane < 16) ? 1.0f : 0.0f;
    for (int t = wave; t < 32 * 32; t += 16) {      // uniform trip count/wave
        const int i0 = (t >> 5) * 16, j0 = (t & 31) * 16;

        // WMMA pairwise sums: D[M][N] = area_i0+M + area_j0+N
        v2f av, bv;
        av.x = sarea[i0 + (lane & 15)] * gate;  av.y = gate;
        bv.x = gate;  bv.y = sarea[j0 + (lane & 15)] * gate;
        v8f acc = {};
        v8f d = __builtin_amdgcn_wmma_f32_16x16x4_f32(
                    false, av, false, bv, (short)0, acc, false, false);

        const int j = j0 + (lane & 15);
        const float jx1 = sbx[j][0], jy1 = sbx[j][1];
        const float jx2 = sbx[j][2], jy2 = sbx[j][3];
        const int ibase = i0 + ((lane >= 16) ? 8 : 0);
        const int word = j0 >> 5;
        const unsigned sh = (unsigned)(j0 & 31);    // 0 or 16
        #pragma unroll
        for (int v = 0; v < 8; ++v) {
            const int i = ibase + v;
            const float ix1 = sbx[i][0], iy1 = sbx[i][1];
            const float ix2 = sbx[i][2], iy2 = sbx[i][3];
            const float dx = fminf(ix2, jx2) - fmaxf(ix1, jx1);
            const float dy = fminf(iy2, jy2) - fmaxf(iy1, jy1);
            const float inter = fmaxf(dx, 0.0f) * fmaxf(dy, 0.0f);
            const float uni = d[v] - inter + 1e-9f;
            const bool sup = inter > IOU_T * uni;   // iou > 0.5
            // ballot is wave-uniform: every lane sees all 32 pair results
            const unsigned bal = (unsigned)__ballot(sup ? 1 : 0);
            // lane 0 owns row i0+v (low 16 bits), lane 16 owns row i0+8+v
            // (high 16 bits); per-lane i already differs between halves.
            const unsigned contrib =
                (lane < 16) ? ((bal & 0xFFFFu) << sh) : ((bal >> 16) << sh);
            if ((lane & 15) == 0) atomicOr(&smask[i][word], contrib);
        }
    }
    __syncthreads();

    // ---- sequential greedy keep: wave 0, 16 mask words across 16 lanes ----
    if (tid < 32) {
        unsigned keepw = 0u;
        for (int i = 0; i < KSEL; ++i) {
            const bool viol = (lane < 16) && ((smask[i][lane & 15] & keepw) != 0u);
            const bool sup  = __any(viol ? 1 : 0) != 0;
            const bool kept = (sscore[i] > 0.0f) && !sup;
            if (kept && lane == (i >> 5)) keepw |= (1u << (i & 31));
        }
        if (lane < 16) skeep[lane] = keepw;
    }
    __syncthreads();

    // ---- target IoU + masked argmax (first-max tie-break) ----
    {
        const float tx1 = target[0], ty1 = target[1];
        const float tx2 = target[2], ty2 = target[3];
        const int   tcls = (int)target[5];
        const float tarea = (tx2 - tx1) * (ty2 - ty1);

        const bool kept = ((skeep[tid >> 5] >> (tid & 31)) & 1u) != 0u;
        const bool tm = kept && (scls[tid] == tcls);
        const float rx1 = sraw[tid][0], ry1 = sraw[tid][1];
        const float rx2 = sraw[tid][2], ry2 = sraw[tid][3];
        const float dx = fminf(rx2, tx2) - fmaxf(rx1, tx1);
        const float dy = fminf(ry2, ty2) - fmaxf(ry1, ty1);
        const float inter = fmaxf(dx, 0.0f) * fmaxf(dy, 0.0f);
        const float areab = (rx2 - rx1) * (ry2 - ry1);
        const float iou = inter / (areab + tarea - inter + 1e-9f);
        redv[tid] = tm ? iou : -1.0f;
        redi[tid] = tid;
    }
    for (int s = KSEL / 2; s > 0; s >>= 1) {
        __syncthreads();
        if (tid < s) {
            const float ov = redv[tid + s]; const int oi = redi[tid + s];
            if (ov > redv[tid] || (ov == redv[tid] && oi < redi[tid])) {
                redv[tid] = ov; redi[tid] = oi;
            }
        }
    }
    __syncthreads();
    if (tid == 0) {
        const float mv = redv[0]; const int j = redi[0];
        out[b] = (mv >= 0.0f) ? sscore[j] * mv : 0.0f;   // mv == iou_t[j]
    }
}

// ============================================================================
// Host launcher
// ============================================================================
extern "C" void kernel_launch(void* const* d_in, const int* in_sizes, int n_in,
                              void* d_out, int out_size, void* d_ws, size_t ws_size,
                              hipStream_t stream) {
    (void)in_sizes; (void)n_in; (void)out_size; (void)ws_size;
    const float* x      = (const float*)d_in[0];   // (16,25200,85) f32
    const float* target = (const float*)d_in[1];   // (6,) f32
    float* out = (float*)d_out;                    // (16,) f32

    unsigned long long* keys   = (unsigned long long*)d_ws;            // B*N u64
    unsigned long long* sorted = keys + (size_t)BATCH * NROWS;         // B*512 u64

    dim3 g1((NROWS + 127) / 128, BATCH);   // 2 x 64-row tiles per block
    score_kernel <<<g1,    256, 0, stream>>>(x, keys);
    select_kernel<<<BATCH, 256, 0, stream>>>(keys, sorted);
    nms_kernel   <<<BATCH, 512, 0, stream>>>(x, target, sorted, out);
}